// MaxPoolingAggregatorPaper_35699768164472
// MI455X (gfx1250) — compile-verified
//
#include <hip/hip_runtime.h>

typedef __attribute__((ext_vector_type(2))) float v2f;
typedef __attribute__((ext_vector_type(4))) float v4f;
typedef __attribute__((ext_vector_type(8))) float v8f;

#define B_SZ  4096
#define NNBR  64
#define DIN   128
#define DHID  128
#define KCOMB 256   // DIN + DHID
#define NOUT  256   // 2*D_OUT

static __device__ __forceinline__ v8f wmma_f32(v2f a, v2f b, v8f c) {
    // D = A(16x4 f32) * B(4x16 f32) + C(16x16 f32)
    return __builtin_amdgcn_wmma_f32_16x16x4_f32(false, a, false, b, (short)0, c,
                                                 false, false);
}

// ---------------------------------------------------------------------------
// Kernel 1: h = relu(neigh[b] @ W_mlp^T + b_mlp), zero masked rows,
//           pooled[b][:] = max over 64 neighbor rows.
// One block per batch row b. 8 waves; wave w owns output columns [16w,16w+16).
// ---------------------------------------------------------------------------
__global__ __launch_bounds__(256)
void pool_mlp_kernel(const float* __restrict__ neigh,
                     const float* __restrict__ Wmlp,
                     const float* __restrict__ bmlp,
                     float* __restrict__ pooled)
{
    __shared__ float sA[NNBR * DIN];   // 32 KB neighbor tile
    __shared__ int   smask[NNBR];

    const int b    = blockIdx.x;
    const int tid  = threadIdx.x;
    const int lane = tid & 31;
    const int wave = tid >> 5;

    const float* gA = neigh + (size_t)b * (NNBR * DIN);

    // Cooperative stage of neigh[b] into LDS: 2048 float4, 8 per thread.
    {
        const v4f* src = (const v4f*)gA;
        v4f*       dst = (v4f*)sA;
        #pragma unroll
        for (int i = 0; i < 8; ++i) {
            int idx = tid + i * 256;
            dst[idx] = src[idx];
        }
    }
    __syncthreads();

    // Padding mask: a neighbor row is padding iff its feature sum is exactly 0
    // (padded rows are all-zero, so any summation order detects them).
    if (tid < NNBR) {
        float s = 0.f;
        for (int k = 0; k < DIN; ++k) s += sA[tid * DIN + k];
        smask[tid] = (s == 0.0f) ? 1 : 0;
    }
    __syncthreads();

    const int halfsel = lane >> 4;      // 0: K={0,1}/M rows 0-7 ; 1: K={2,3}/M rows 8-15
    const int lmod    = lane & 15;
    const int n       = wave * 16 + lmod;   // this lane's output column

    const float* wrow = Wmlp + (size_t)n * DIN;   // B(k,n) = W[n][k], contiguous in k

    v8f acc0 = {}; v8f acc1 = {}; v8f acc2 = {}; v8f acc3 = {};

    for (int k = 0; k < DIN; k += 4) {
        const int kb = k + halfsel * 2;
        v2f bf = *(const v2f*)(wrow + kb);                       // B fragment (reused x4)
        v2f a0 = *(const v2f*)(&sA[( 0 + lmod) * DIN + kb]);     // A frags, 4 M-tiles
        v2f a1 = *(const v2f*)(&sA[(16 + lmod) * DIN + kb]);
        v2f a2 = *(const v2f*)(&sA[(32 + lmod) * DIN + kb]);
        v2f a3 = *(const v2f*)(&sA[(48 + lmod) * DIN + kb]);
        acc0 = wmma_f32(a0, bf, acc0);
        acc1 = wmma_f32(a1, bf, acc1);
        acc2 = wmma_f32(a2, bf, acc2);
        acc3 = wmma_f32(a3, bf, acc3);
    }

    // Epilogue: bias + relu + mask, column max over the 64 neighbor rows.
    const float bias = bmlp[n];
    float cmax = 0.0f;  // relu floor == masked value, consistent with reference
    #pragma unroll
    for (int r = 0; r < 8; ++r) {
        const int mr = r + halfsel * 8;   // row within 16-row tile (C/D layout)
        float v0 = fmaxf(acc0[r] + bias, 0.f); if (smask[ 0 + mr]) v0 = 0.f;
        float v1 = fmaxf(acc1[r] + bias, 0.f); if (smask[16 + mr]) v1 = 0.f;
        float v2 = fmaxf(acc2[r] + bias, 0.f); if (smask[32 + mr]) v2 = 0.f;
        float v3 = fmaxf(acc3[r] + bias, 0.f); if (smask[48 + mr]) v3 = 0.f;
        cmax = fmaxf(cmax, fmaxf(fmaxf(v0, v1), fmaxf(v2, v3)));
    }
    // Column n lives in lanes l and l+16 (different row halves): merge halves.
    cmax = fmaxf(cmax, __shfl_xor(cmax, 16, 32));
    if (lmod == lane)  // lanes 0..15 write
        pooled[(size_t)b * DHID + n] = cmax;
}

// ---------------------------------------------------------------------------
// Kernel 2: out = relu(concat(input, pooled) @ W_comb^T + b_comb)
// One block per 16-row M-tile; 8 waves x 2 N-tiles each cover 256 columns.
// K=256 splits at 128: first half reads input, second half reads pooled.
// ---------------------------------------------------------------------------
__global__ __launch_bounds__(256)
void combine_kernel(const float* __restrict__ input,
                    const float* __restrict__ pooled,
                    const float* __restrict__ Wcomb,
                    const float* __restrict__ bcomb,
                    float* __restrict__ out)
{
    const int m0   = blockIdx.x * 16;
    const int tid  = threadIdx.x;
    const int lane = tid & 31;
    const int wave = tid >> 5;

    const int halfsel = lane >> 4;
    const int lmod    = lane & 15;
    const int m       = m0 + lmod;

    const int na = wave * 32 + lmod;        // first  N-tile column
    const int nb = na + 16;                 // second N-tile column
    const float* wra = Wcomb + (size_t)na * KCOMB;
    const float* wrb = Wcomb + (size_t)nb * KCOMB;

    v8f accA = {}; v8f accB = {};

    // K in [0,128): A rows come from `input`
    const float* arow0 = input + (size_t)m * DIN;
    for (int k = 0; k < DIN; k += 4) {
        const int kb = k + halfsel * 2;
        v2f a  = *(const v2f*)(arow0 + kb);
        v2f b0 = *(const v2f*)(wra + kb);
        v2f b1 = *(const v2f*)(wrb + kb);
        accA = wmma_f32(a, b0, accA);
        accB = wmma_f32(a, b1, accB);
    }
    // K in [128,256): A rows come from `pooled`
    const float* arow1 = pooled + (size_t)m * DHID;
    for (int k = 0; k < DHID; k += 4) {
        const int kb = k + halfsel * 2;
        v2f a  = *(const v2f*)(arow1 + kb);
        v2f b0 = *(const v2f*)(wra + DIN + kb);
        v2f b1 = *(const v2f*)(wrb + DIN + kb);
        accA = wmma_f32(a, b0, accA);
        accB = wmma_f32(a, b1, accB);
    }

    const float biasA = bcomb[na];
    const float biasB = bcomb[nb];
    #pragma unroll
    for (int r = 0; r < 8; ++r) {
        const int mr = m0 + r + halfsel * 8;
        out[(size_t)mr * NOUT + na] = fmaxf(accA[r] + biasA, 0.f);
        out[(size_t)mr * NOUT + nb] = fmaxf(accB[r] + biasB, 0.f);
    }
}

// ---------------------------------------------------------------------------
extern "C" void kernel_launch(void* const* d_in, const int* in_sizes, int n_in,
                              void* d_out, int out_size, void* d_ws, size_t ws_size,
                              hipStream_t stream)
{
    const float* input = (const float*)d_in[0];  // [4096,128]
    const float* neigh = (const float*)d_in[1];  // [4096,64,128]
    const float* Wmlp  = (const float*)d_in[2];  // [128,128]
    const float* bmlp  = (const float*)d_in[3];  // [128]
    const float* Wcomb = (const float*)d_in[4];  // [256,256]
    const float* bcomb = (const float*)d_in[5];  // [256]
    float* out    = (float*)d_out;               // [4096,256]
    float* pooled = (float*)d_ws;                // [4096,128] scratch (2 MB)

    pool_mlp_kernel<<<B_SZ, 256, 0, stream>>>(neigh, Wmlp, bmlp, pooled);
    combine_kernel<<<B_SZ / 16, 256, 0, stream>>>(input, pooled, Wcomb, bcomb, out);
}